// PetriNetAlignmentPredictor_84945863180732
// MI455X (gfx1250) — compile-verified
//
#include <hip/hip_runtime.h>
#include <math.h>

// ---------------------------------------------------------------------------
// PetriNetAlignmentPredictor forward, MI455X (gfx1250, wave32).
//
// Phases:
//   1) proj:   v_cur/v_tgt = {v_src,v_target} @ normalize(emb)   (128MB stream)
//   2) gens:   C[512x256] = coeffs[512x120] x bases[120x256] via
//              V_WMMA_F32_16X16X4_F32 (exact: products are coeff * {0,+-1})
//   3) steps:  64 serial steps in ONE workgroup (latency-bound):
//              MLP matvecs -> argmax(logits+gumbel) -> omega=gens[t*]
//              -> expm(omega)@v via 24-term Taylor matvec -> renormalize
//   4) softmax over 2M scores = emb_norm . v_fin  (128MB stream + 3x8MB)
// All reductions are fixed-shape two-pass => bitwise deterministic.
// ---------------------------------------------------------------------------

typedef __attribute__((ext_vector_type(2))) float v2f;
typedef __attribute__((ext_vector_type(8))) float v8f;

#define N_DIM   16
#define NUM_TR  512
#define NUM_GEN 120
#define MLP_H1  256
#define MLP_H2  128
#define TAYLOR_K 24
#define PROJ_BLOCKS 1024
#define PROJ_THREADS 256

// workspace layout (float offsets)
#define WS_VCUR  0                          // 32: vcur[16], vtgt[16]
#define WS_VFIN  32                         // 16
#define WS_GMAX  48                         // 1
#define WS_GINV  49                         // 1 (1/sum)
#define WS_PART  64                         // PROJ_BLOCKS*32 partials; reused for blk max/sum
#define WS_GENS  (64 + PROJ_BLOCKS * 32)    // 512*256 floats (~512KB)

// ---------------------------------------------------------------------------
// Phase 1: partial projections. Each block -> 32 partial sums.
__global__ __launch_bounds__(PROJ_THREADS)
void proj_partial_kernel(const float* __restrict__ v_src,
                         const float* __restrict__ v_tgt,
                         const float* __restrict__ emb,
                         float* __restrict__ partials, int num_m)
{
    __shared__ float red[PROJ_THREADS * 32];
    float as[N_DIM], at[N_DIM];
#pragma unroll
    for (int i = 0; i < N_DIM; ++i) { as[i] = 0.f; at[i] = 0.f; }

    const int stride = blockDim.x * gridDim.x;
    for (int m = blockIdx.x * blockDim.x + threadIdx.x; m < num_m; m += stride) {
        const float4* row4 = reinterpret_cast<const float4*>(emb + (size_t)m * N_DIM);
        if (m + stride < num_m)
            __builtin_prefetch(emb + (size_t)(m + stride) * N_DIM, 0, 2);
        float4 r0 = row4[0], r1 = row4[1], r2 = row4[2], r3 = row4[3];
        float row[N_DIM] = { r0.x, r0.y, r0.z, r0.w, r1.x, r1.y, r1.z, r1.w,
                             r2.x, r2.y, r2.z, r2.w, r3.x, r3.y, r3.z, r3.w };
        float ss = 0.f;
#pragma unroll
        for (int i = 0; i < N_DIM; ++i) ss = fmaf(row[i], row[i], ss);
        float inv = 1.0f / sqrtf(ss);
        float cs = v_src[m] * inv, ct = v_tgt[m] * inv;
#pragma unroll
        for (int i = 0; i < N_DIM; ++i) {
            as[i] = fmaf(cs, row[i], as[i]);
            at[i] = fmaf(ct, row[i], at[i]);
        }
    }
#pragma unroll
    for (int i = 0; i < N_DIM; ++i) {
        red[threadIdx.x * 32 + i]         = as[i];
        red[threadIdx.x * 32 + N_DIM + i] = at[i];
    }
    __syncthreads();
    if (threadIdx.x < 32) {
        float s = 0.f;
        for (int t = 0; t < PROJ_THREADS; ++t) s += red[t * 32 + threadIdx.x];
        partials[blockIdx.x * 32 + threadIdx.x] = s;
    }
}

__global__ void proj_reduce_kernel(const float* __restrict__ partials,
                                   float* __restrict__ vout)
{
    int c = threadIdx.x;
    if (c < 32) {
        float s = 0.f;
        for (int b = 0; b < PROJ_BLOCKS; ++b) s += partials[b * 32 + c];
        vout[c] = s;
    }
}

// ---------------------------------------------------------------------------
// Phase 2: gens GEMM via V_WMMA_F32_16X16X4_F32.
// One wave per 16x16 tile of C[512x256]; K=120 in steps of 4.
__global__ __launch_bounds__(256)
void gens_wmma_kernel(const float* __restrict__ coeffs,  // [512,120]
                      const float* __restrict__ bases,   // [120,256]
                      float* __restrict__ gens)          // [512,256]
{
    const int wave = blockIdx.x * (blockDim.x >> 5) + (threadIdx.x >> 5); // 0..511
    const int lane = threadIdx.x & 31;
    const int tm = wave >> 4;      // 32 row tiles
    const int tn = wave & 15;      // 16 col tiles
    const int half = lane >> 4;    // 0: K0/K1 ; 1: K2/K3
    const int l = lane & 15;

    v8f acc = {};
    for (int k = 0; k < NUM_GEN; k += 4) {
        const int ka = k + half * 2;
        v2f a, b;
        // A fragment: 16x4, lanes 0-15 rows M=0..15 (K0,K1), lanes 16-31 (K2,K3)
        a.x = coeffs[(tm * 16 + l) * NUM_GEN + ka + 0];
        a.y = coeffs[(tm * 16 + l) * NUM_GEN + ka + 1];
        // B fragment: 4x16, columns across lanes, K down VGPRs, halves split K
        b.x = bases[(ka + 0) * 256 + tn * 16 + l];
        b.y = bases[(ka + 1) * 256 + tn * 16 + l];
        acc = __builtin_amdgcn_wmma_f32_16x16x4_f32(false, a, false, b,
                                                    (short)0, acc, false, false);
    }
    // C/D layout: VGPR r -> lanes0-15: M=r, lanes16-31: M=r+8
#pragma unroll
    for (int r = 0; r < 8; ++r) {
        int row = tm * 16 + r + half * 8;
        int col = tn * 16 + l;
        gens[row * 256 + col] = acc[r];
    }
}

// ---------------------------------------------------------------------------
// Phase 3: 64 serial steps, one workgroup of 512 threads.
__global__ __launch_bounds__(512)
void steps_kernel(const float* __restrict__ w1, const float* __restrict__ b1,
                  const float* __restrict__ w2, const float* __restrict__ b2,
                  const float* __restrict__ w3, const float* __restrict__ b3,
                  const float* __restrict__ gumbel,
                  const float* __restrict__ gens,
                  const float* __restrict__ vio,      // ws: vcur[16], vtgt[16]
                  const int*   __restrict__ num_steps_p,
                  float* __restrict__ logits_out,     // d_out + 2M
                  float* __restrict__ vfin)           // ws
{
    __shared__ float x[2 * N_DIM];
    __shared__ float h1[MLP_H1];
    __shared__ float h2[MLP_H2];
    __shared__ float rval[NUM_TR];
    __shared__ int   ridx[NUM_TR];
    __shared__ float omega[N_DIM * N_DIM];
    __shared__ float vterm[N_DIM], vacc[N_DIM], vvec[N_DIM], vtg[N_DIM];
    __shared__ float sinv;

    const int tid = threadIdx.x;
    const int ns = *num_steps_p;

    if (tid < N_DIM) { vvec[tid] = vio[tid]; vtg[tid] = vio[N_DIM + tid]; }
    __syncthreads();

    for (int s = 0; s < ns; ++s) {
        if (tid < N_DIM) { x[tid] = vvec[tid]; x[N_DIM + tid] = vtg[tid]; }
        __syncthreads();
        // layer 1: [1,32] @ [32,256] + relu
        if (tid < MLP_H1) {
            float a = b1[tid];
#pragma unroll
            for (int i = 0; i < 2 * N_DIM; ++i)
                a = fmaf(x[i], w1[i * MLP_H1 + tid], a);
            h1[tid] = fmaxf(a, 0.f);
        }
        __syncthreads();
        // layer 2: [1,256] @ [256,128] + relu
        if (tid < MLP_H2) {
            float a = b2[tid];
            for (int i = 0; i < MLP_H1; ++i)
                a = fmaf(h1[i], w2[i * MLP_H2 + tid], a);
            h2[tid] = fmaxf(a, 0.f);
        }
        __syncthreads();
        // layer 3: [1,128] @ [128,512]; emit logits, perturb with gumbel
        {
            float a = b3[tid];
            for (int i = 0; i < MLP_H2; ++i)
                a = fmaf(h2[i], w3[i * NUM_TR + tid], a);
            logits_out[s * NUM_TR + tid] = a;
            rval[tid] = a + gumbel[s * NUM_TR + tid];
            ridx[tid] = tid;
        }
        // argmax (first-max tie-break matches jnp.argmax)
        for (int off = NUM_TR / 2; off > 0; off >>= 1) {
            __syncthreads();
            if (tid < off) {
                float va = rval[tid], vb = rval[tid + off];
                int ia = ridx[tid], ib = ridx[tid + off];
                if (vb > va || (vb == va && ib < ia)) { rval[tid] = vb; ridx[tid] = ib; }
            }
        }
        __syncthreads();
        const int tstar = ridx[0];
        if (tid < N_DIM * N_DIM) omega[tid] = gens[tstar * (N_DIM * N_DIM) + tid];
        if (tid < N_DIM) { vacc[tid] = vvec[tid]; vterm[tid] = vvec[tid]; }
        __syncthreads();
        // v2 = expm(omega) @ v via Taylor matvec recurrence (||omega|| ~ 1)
        for (int k = 1; k <= TAYLOR_K; ++k) {
            float nt = 0.f;
            if (tid < N_DIM) {
#pragma unroll
                for (int j = 0; j < N_DIM; ++j)
                    nt = fmaf(omega[tid * N_DIM + j], vterm[j], nt);
                nt /= (float)k;
            }
            __syncthreads();
            if (tid < N_DIM) { vterm[tid] = nt; vacc[tid] += nt; }
            __syncthreads();
        }
        if (tid == 0) {
            float ss = 0.f;
            for (int i = 0; i < N_DIM; ++i) ss = fmaf(vacc[i], vacc[i], ss);
            sinv = 1.0f / sqrtf(ss);
        }
        __syncthreads();
        if (tid < N_DIM) vvec[tid] = vacc[tid] * sinv;
        __syncthreads();
    }
    if (tid < N_DIM) vfin[tid] = vvec[tid];
}

// ---------------------------------------------------------------------------
// Phase 4: scores = emb_norm . v_fin  -> d_out; per-block max -> ws.
__global__ __launch_bounds__(PROJ_THREADS)
void scores_kernel(const float* __restrict__ emb,
                   const float* __restrict__ vfin,
                   float* __restrict__ scores,
                   float* __restrict__ blkmax, int num_m)
{
    __shared__ float v[N_DIM];
    __shared__ float red[PROJ_THREADS];
    if (threadIdx.x < N_DIM) v[threadIdx.x] = vfin[threadIdx.x];
    __syncthreads();
    float mx = -INFINITY;
    const int stride = blockDim.x * gridDim.x;
    for (int m = blockIdx.x * blockDim.x + threadIdx.x; m < num_m; m += stride) {
        const float4* row4 = reinterpret_cast<const float4*>(emb + (size_t)m * N_DIM);
        if (m + stride < num_m)
            __builtin_prefetch(emb + (size_t)(m + stride) * N_DIM, 0, 2);
        float4 r0 = row4[0], r1 = row4[1], r2 = row4[2], r3 = row4[3];
        float row[N_DIM] = { r0.x, r0.y, r0.z, r0.w, r1.x, r1.y, r1.z, r1.w,
                             r2.x, r2.y, r2.z, r2.w, r3.x, r3.y, r3.z, r3.w };
        float ss = 0.f, d = 0.f;
#pragma unroll
        for (int i = 0; i < N_DIM; ++i) {
            ss = fmaf(row[i], row[i], ss);
            d  = fmaf(row[i], v[i], d);
        }
        float sc = d * (1.0f / sqrtf(ss));
        scores[m] = sc;
        mx = fmaxf(mx, sc);
    }
    red[threadIdx.x] = mx;
    for (int off = PROJ_THREADS / 2; off > 0; off >>= 1) {
        __syncthreads();
        if (threadIdx.x < off)
            red[threadIdx.x] = fmaxf(red[threadIdx.x], red[threadIdx.x + off]);
    }
    if (threadIdx.x == 0) blkmax[blockIdx.x] = red[0];
}

__global__ void reduce_max_kernel(const float* __restrict__ blk, float* __restrict__ gmax)
{
    __shared__ float red[256];
    float m = -INFINITY;
    for (int i = threadIdx.x; i < PROJ_BLOCKS; i += 256) m = fmaxf(m, blk[i]);
    red[threadIdx.x] = m;
    for (int off = 128; off > 0; off >>= 1) {
        __syncthreads();
        if (threadIdx.x < off)
            red[threadIdx.x] = fmaxf(red[threadIdx.x], red[threadIdx.x + off]);
    }
    if (threadIdx.x == 0) *gmax = red[0];
}

__global__ __launch_bounds__(PROJ_THREADS)
void expsum_kernel(float* __restrict__ scores, const float* __restrict__ gmax,
                   float* __restrict__ blksum, int num_m)
{
    __shared__ float red[PROJ_THREADS];
    const float g = *gmax;
    float s = 0.f;
    const int stride = blockDim.x * gridDim.x;
    for (int m = blockIdx.x * blockDim.x + threadIdx.x; m < num_m; m += stride) {
        float e = expf(scores[m] - g);
        scores[m] = e;
        s += e;
    }
    red[threadIdx.x] = s;
    for (int off = PROJ_THREADS / 2; off > 0; off >>= 1) {
        __syncthreads();
        if (threadIdx.x < off) red[threadIdx.x] += red[threadIdx.x + off];
    }
    if (threadIdx.x == 0) blksum[blockIdx.x] = red[0];
}

__global__ void reduce_sum_kernel(const float* __restrict__ blk, float* __restrict__ ginv)
{
    __shared__ float red[256];
    float s = 0.f;
    for (int i = threadIdx.x; i < PROJ_BLOCKS; i += 256) s += blk[i];
    red[threadIdx.x] = s;
    for (int off = 128; off > 0; off >>= 1) {
        __syncthreads();
        if (threadIdx.x < off) red[threadIdx.x] += red[threadIdx.x + off];
    }
    if (threadIdx.x == 0) *ginv = 1.0f / red[0];
}

__global__ __launch_bounds__(PROJ_THREADS)
void scale_kernel(float* __restrict__ scores, const float* __restrict__ ginv, int num_m)
{
    const float inv = *ginv;
    const int stride = blockDim.x * gridDim.x;
    for (int m = blockIdx.x * blockDim.x + threadIdx.x; m < num_m; m += stride)
        scores[m] *= inv;
}

// ---------------------------------------------------------------------------
extern "C" void kernel_launch(void* const* d_in, const int* in_sizes, int n_in,
                              void* d_out, int out_size, void* d_ws, size_t ws_size,
                              hipStream_t stream)
{
    const float* v_src  = (const float*)d_in[0];
    const float* v_tgt  = (const float*)d_in[1];
    const float* emb    = (const float*)d_in[2];
    const float* coeffs = (const float*)d_in[3];
    const float* bases  = (const float*)d_in[4];
    const float* w1 = (const float*)d_in[5];
    const float* b1 = (const float*)d_in[6];
    const float* w2 = (const float*)d_in[7];
    const float* b2 = (const float*)d_in[8];
    const float* w3 = (const float*)d_in[9];
    const float* b3 = (const float*)d_in[10];
    const float* gumbel = (const float*)d_in[11];
    const int* num_steps = (const int*)d_in[12];

    const int num_m = in_sizes[0];          // 2,000,000
    float* out = (float*)d_out;
    float* pred = out;                      // [num_m]
    float* logits_out = out + num_m;        // [64*512]
    float* ws = (float*)d_ws;

    // 1) projections
    proj_partial_kernel<<<PROJ_BLOCKS, PROJ_THREADS, 0, stream>>>(
        v_src, v_tgt, emb, ws + WS_PART, num_m);
    proj_reduce_kernel<<<1, 32, 0, stream>>>(ws + WS_PART, ws + WS_VCUR);

    // 2) gens GEMM (WMMA): 512 tiles -> 64 blocks x 8 waves
    gens_wmma_kernel<<<64, 256, 0, stream>>>(coeffs, bases, ws + WS_GENS);

    // 3) serial step loop
    steps_kernel<<<1, 512, 0, stream>>>(w1, b1, w2, b2, w3, b3, gumbel,
                                        ws + WS_GENS, ws + WS_VCUR, num_steps,
                                        logits_out, ws + WS_VFIN);

    // 4) softmax over 2M
    scores_kernel<<<PROJ_BLOCKS, PROJ_THREADS, 0, stream>>>(
        emb, ws + WS_VFIN, pred, ws + WS_PART, num_m);
    reduce_max_kernel<<<1, 256, 0, stream>>>(ws + WS_PART, ws + WS_GMAX);
    expsum_kernel<<<PROJ_BLOCKS, PROJ_THREADS, 0, stream>>>(
        pred, ws + WS_GMAX, ws + WS_PART, num_m);
    reduce_sum_kernel<<<1, 256, 0, stream>>>(ws + WS_PART, ws + WS_GINV);
    scale_kernel<<<PROJ_BLOCKS, PROJ_THREADS, 0, stream>>>(pred, ws + WS_GINV, num_m);
}